// GraphNodeEncoder_79972291052144
// MI455X (gfx1250) — compile-verified
//
#include <hip/hip_runtime.h>
#include <hip/hip_bf16.h>

typedef __attribute__((ext_vector_type(2))) float v2f;
typedef __attribute__((ext_vector_type(4))) float f4;
typedef __attribute__((ext_vector_type(8))) float v8f;

#define NN 8192
#define DD 64
#define BM 64          // graph rows per block
#define BK 64          // k-tile
#define LDT 68         // LDS row stride (dwords): b64 pair reads are bank-conflict-free

__device__ __forceinline__ v8f wmma4(v2f a, v2f b, v8f c) {
  // D(16x16 f32) = A(16x4 f32) * B(4x16 f32) + C
  return __builtin_amdgcn_wmma_f32_16x16x4_f32(false, a, false, b, (short)0, c,
                                               false, false);
}

// Async 16B global->LDS copy, GVS addressing: mem = saddr + voff. ASYNCcnt-tracked.
__device__ __forceinline__ void async_b128(unsigned lds, unsigned voff,
                                           const void* sbase) {
  asm volatile("global_load_async_to_lds_b128 %0, %1, %2"
               :: "v"(lds), "v"(voff), "s"(sbase)
               : "memory");
}

__global__ __launch_bounds__(256) void k_dis(const float* __restrict__ deg,
                                             float* __restrict__ dis) {
  int i = blockIdx.x * 256 + threadIdx.x;
  dis[i] = rsqrtf(deg[i] + 1.0f);
}

// xsT[c][r] = x[r][c] * dis[r]  (feature-major), LDS-tiled transpose.
__global__ __launch_bounds__(256) void k_prescaleT(const float* __restrict__ x,
                                                   const float* __restrict__ dis,
                                                   float* __restrict__ xsT) {
  __shared__ float t[64 * 65];
  const int rb = blockIdx.x * 64;
  const int tid = threadIdx.x;
#pragma unroll
  for (int i = 0; i < 16; ++i) {
    int lin = i * 256 + tid;
    int r = lin >> 6, c = lin & 63;
    t[c * 65 + r] = x[(size_t)(rb + r) * DD + c] * dis[rb + r];
  }
  __syncthreads();
#pragma unroll
  for (int i = 0; i < 16; ++i) {
    int lin = i * 256 + tid;
    int c = lin >> 6, r = lin & 63;
    xsT[(size_t)c * NN + rb + r] = t[c * 65 + r];
  }
}

// One GCN layer: agg^T = xsT * adj^T via WMMA, fused W-GEMM epilogue.
// outp: feature-major xs_next (=relu(h)*dis) if !LAST, row-major final if LAST.
template <bool LAST>
__global__ __launch_bounds__(256) void gcn_layer(
    const float* __restrict__ adj, const float* __restrict__ xsT,
    const float* __restrict__ dis, const float* __restrict__ Wm,
    const float* __restrict__ bias, const float* __restrict__ xres,
    const float* __restrict__ gamma, const float* __restrict__ beta,
    float* __restrict__ outp) {
  __shared__ float lAdj[2][BM * LDT];  // adj tiles [graph-row][k]; [0] reused as staging
  __shared__ float lXs[2][DD * LDT];   // xsT tiles [feature][k]; [0] reused as reduce buf
  __shared__ float lWT[DD * LDT];      // W^T [out-feat][in-feat], resident

  const int tid = threadIdx.x;
  const int w = tid >> 5;      // wave 0..7
  const int lane = tid & 31;
  const int ln = lane & 15;
  const int hi = lane >> 4;    // selects K pair / M+8 half
  const int rt = w & 3;        // graph-row tile within block
  const int kh = w >> 2;       // k-half
  const int rb = blockIdx.x * BM;

  // Stage W^T once: lWT[m][k] = W[k][m]
#pragma unroll
  for (int i = 0; i < 16; ++i) {
    int m = tid & 63, k = (tid >> 6) * 16 + i;
    lWT[m * LDT + k] = Wm[k * DD + m];
  }

  // Loop-invariant per-lane pieces for the async copies.
  unsigned voff[4], ldsAdj[2][4], ldsXs[2][4];
#pragma unroll
  for (int i = 0; i < 4; ++i) {
    int lin = i * 256 + tid;
    int r = lin >> 4, c4 = (lin & 15) << 2;
    voff[i] = (unsigned)((r * NN + c4) * 4);
#pragma unroll
    for (int p = 0; p < 2; ++p) {
      ldsAdj[p][i] = (unsigned)(uintptr_t)&lAdj[p][r * LDT + c4];
      ldsXs[p][i] = (unsigned)(uintptr_t)&lXs[p][r * LDT + c4];
    }
  }
  const float* adjBase = adj + (size_t)rb * NN;  // uniform; + k per tile

  auto issue_tile = [&](int k, int buf) {
#pragma unroll
    for (int i = 0; i < 4; ++i) {
      async_b128(ldsAdj[buf][i], voff[i], adjBase + k);
      async_b128(ldsXs[buf][i], voff[i], xsT + k);
    }
  };

  v8f acc[4] = {};  // acc[t]: agg^T tile, M = feat 16t.., N = graph rows 16rt..
  const int NT = NN / BK;

  issue_tile(0, 0);
  for (int it = 0; it < NT; ++it) {
    const int p = it & 1;
    if (it + 1 < NT) {
      issue_tile((it + 1) * BK, 1 - p);                 // prefetch next tile
      asm volatile("s_wait_asynccnt 0x8" ::: "memory"); // tile `it` landed (in-order)
    } else {
      asm volatile("s_wait_asynccnt 0x0" ::: "memory");
    }
    __syncthreads();
    const float* bAdj = &lAdj[p][0];
    const float* bXs = &lXs[p][0];
    const int kb = kh * 32;
#pragma unroll
    for (int kk = 0; kk < 32; kk += 4) {
      const int kq = kb + kk + 2 * hi;
      v2f b = *(const v2f*)&bAdj[(rt * 16 + ln) * LDT + kq];  // adj^T fragment
#pragma unroll
      for (int t = 0; t < 4; ++t) {
        v2f a = *(const v2f*)&bXs[(t * 16 + ln) * LDT + kq];  // xs^T fragment
        acc[t] = wmma4(a, b, acc[t]);
      }
    }
    __syncthreads();
  }

  // Reduce K-halves: waves 4..7 dump partials, waves 0..3 accumulate.
  if (w >= 4) {
    float* red = &lXs[0][0];
#pragma unroll
    for (int t = 0; t < 4; ++t)
#pragma unroll
      for (int j = 0; j < 8; ++j)
        red[((w - 4) * 16 + ln) * LDT + t * 16 + hi * 8 + j] = acc[t][j];
  }
  __syncthreads();

  if (w < 4) {
    float* red = &lXs[0][0];
#pragma unroll
    for (int t = 0; t < 4; ++t)
#pragma unroll
      for (int j = 0; j < 8; ++j)
        acc[t][j] += red[(w * 16 + ln) * LDT + t * 16 + hi * 8 + j];

    const int rowBase = rb + 16 * rt;
    const float dsc = dis[rowBase + ln];  // one graph row per lane

    // Stage scaled agg^T as [row][feature] for the W GEMM (wave-private region).
    float* stg = &lAdj[0][(w * 16) * LDT];
#pragma unroll
    for (int t = 0; t < 4; ++t)
#pragma unroll
      for (int j = 0; j < 8; ++j)
        stg[ln * LDT + t * 16 + hi * 8 + j] = acc[t][j] * dsc;

    // h^T = W^T * agg_scaled^T : both operands natural b64 fragments.
    v8f acc2[4] = {};
#pragma unroll
    for (int kk = 0; kk < DD; kk += 4) {
      const int kq = kk + 2 * hi;
      v2f b2 = *(const v2f*)&stg[ln * LDT + kq];
#pragma unroll
      for (int t = 0; t < 4; ++t) {
        v2f a2 = *(const v2f*)&lWT[(t * 16 + ln) * LDT + kq];
        acc2[t] = wmma4(a2, b2, acc2[t]);
      }
    }

    if (!LAST) {
      // relu(h + b) * dis, stored feature-major (the xsT layout): coalesced b32s.
#pragma unroll
      for (int t = 0; t < 4; ++t) {
        f4 b0 = *(const f4*)&bias[t * 16 + hi * 8];
        f4 b1 = *(const f4*)&bias[t * 16 + hi * 8 + 4];
#pragma unroll
        for (int j = 0; j < 8; ++j) {
          float v = fmaxf(acc2[t][j] + (j < 4 ? b0[j] : b1[j - 4]), 0.0f) * dsc;
          outp[(size_t)(t * 16 + hi * 8 + j) * NN + rowBase + ln] = v;
        }
      }
    } else {
      // h + b + x, then LayerNorm over D=64 (row lives in lanes {ln, ln+16}).
      float val[4][8];
      const float* xrow = &xres[(size_t)(rowBase + ln) * DD];
#pragma unroll
      for (int t = 0; t < 4; ++t) {
        f4 b0 = *(const f4*)&bias[t * 16 + hi * 8];
        f4 b1 = *(const f4*)&bias[t * 16 + hi * 8 + 4];
        f4 x0 = *(const f4*)&xrow[t * 16 + hi * 8];
        f4 x1 = *(const f4*)&xrow[t * 16 + hi * 8 + 4];
#pragma unroll
        for (int j = 0; j < 8; ++j)
          val[t][j] = acc2[t][j] + (j < 4 ? b0[j] + x0[j] : b1[j - 4] + x1[j - 4]);
      }
      float pr = 0.0f, q = 0.0f;
#pragma unroll
      for (int t = 0; t < 4; ++t)
#pragma unroll
        for (int j = 0; j < 8; ++j) {
          float v = val[t][j];
          pr += v;
          q += v * v;
        }
      pr += __shfl_xor(pr, 16, 32);
      q += __shfl_xor(q, 16, 32);
      float mu = pr * (1.0f / 64.0f);
      float var = q * (1.0f / 64.0f) - mu * mu;
      float rstd = rsqrtf(var + 1e-5f);
      // Normalize into LDS staging, then store 16x64 row-major coalesced (b128).
#pragma unroll
      for (int t = 0; t < 4; ++t) {
        f4 g0 = *(const f4*)&gamma[t * 16 + hi * 8];
        f4 g1 = *(const f4*)&gamma[t * 16 + hi * 8 + 4];
        f4 e0 = *(const f4*)&beta[t * 16 + hi * 8];
        f4 e1 = *(const f4*)&beta[t * 16 + hi * 8 + 4];
#pragma unroll
        for (int j = 0; j < 8; ++j) {
          float g = (j < 4 ? g0[j] : g1[j - 4]);
          float e = (j < 4 ? e0[j] : e1[j - 4]);
          stg[ln * LDT + t * 16 + hi * 8 + j] = (val[t][j] - mu) * rstd * g + e;
        }
      }
      // Wave-private LDS region; same-wave DS ops stay in order, no barrier needed.
#pragma unroll
      for (int pi = 0; pi < 8; ++pi) {
        int lin = pi * 128 + lane * 4;
        int r = lin >> 6, c = lin & 63;
        f4 vv = *(const f4*)&stg[r * LDT + c];
        *(f4*)&outp[(size_t)(rowBase + r) * DD + c] = vv;
      }
    }
  }
}

extern "C" void kernel_launch(void* const* d_in, const int* in_sizes, int n_in,
                              void* d_out, int out_size, void* d_ws, size_t ws_size,
                              hipStream_t stream) {
  const float* x      = (const float*)d_in[0];
  const float* adj    = (const float*)d_in[1];
  const float* degree = (const float*)d_in[2];
  const float* W0     = (const float*)d_in[3];
  const float* b0     = (const float*)d_in[4];
  const float* W1     = (const float*)d_in[5];
  const float* b1     = (const float*)d_in[6];
  const float* gamma  = (const float*)d_in[7];
  const float* beta   = (const float*)d_in[8];
  float* out = (float*)d_out;

  float* dis  = (float*)d_ws;            // [NN]
  float* xs0T = dis + NN;                // [DD][NN]  (x*dis)^T
  float* xs1T = xs0T + (size_t)NN * DD;  // [DD][NN]  (relu(h1)*dis)^T

  k_dis<<<NN / 256, 256, 0, stream>>>(degree, dis);
  k_prescaleT<<<NN / 64, 256, 0, stream>>>(x, dis, xs0T);
  gcn_layer<false><<<NN / BM, 256, 0, stream>>>(adj, xs0T, dis, W0, b0, x, gamma, beta, xs1T);
  gcn_layer<true><<<NN / BM, 256, 0, stream>>>(adj, xs1T, dis, W1, b1, x, gamma, beta, out);
}